// EfficientDetONNXModel_8117488190244
// MI455X (gfx1250) — compile-verified
//
#include <hip/hip_runtime.h>
#include <stdint.h>

// ---------------- problem constants (match reference) ----------------
#define NCLS        80
#define MAX_DET     100
#define CAP         2048          // candidate buffer (expected ~1100 pass thr)
#define SCORE_THR   0.2f
#define IOU_THR     0.2f
#define IMGF        768.0f
#define NMS_THREADS 1024

// ---------------- optional CDNA5 async global->LDS path ----------------
#if defined(__has_builtin)
#if __has_builtin(__builtin_amdgcn_global_load_async_to_lds_b32) && \
    __has_builtin(__builtin_amdgcn_global_load_async_to_lds_b128) && \
    __has_builtin(__builtin_amdgcn_s_wait_asynccnt)
#define HAVE_ASYNC_LDS 1
#endif
#endif
#ifndef HAVE_ASYNC_LDS
#define HAVE_ASYNC_LDS 0
#endif

typedef int v4i __attribute__((__vector_size__(4 * sizeof(int))));
typedef __attribute__((address_space(1))) int  g_i32;
typedef __attribute__((address_space(1))) v4i  g_v4i;
typedef __attribute__((address_space(3))) int  l_i32;
typedef __attribute__((address_space(3))) v4i  l_v4i;

// ---------------- kernel 0: zero outputs + reset candidate counter ----
__global__ void init_kernel(float* __restrict__ out, int out_size, int* __restrict__ cnt) {
    int t = blockIdx.x * blockDim.x + threadIdx.x;
    if (t < out_size) out[t] = 0.0f;
    if (t == 0) *cnt = 0;
}

// ---------------- kernel 1: sigmoid-max + threshold + decode + compact -
// One wave32 per anchor. 80 logits are read as 40 float2 (fully coalesced).
__global__ __launch_bounds__(256)
void score_decode_kernel(const float* __restrict__ cls,   // [A,80]
                         const float* __restrict__ reg,   // [A,4]
                         const float* __restrict__ anc,   // [A,4]
                         int A,
                         int* __restrict__ cnt,
                         float* __restrict__ cscore,      // [CAP]
                         float* __restrict__ cbox,        // [CAP,4] x1y1x2y2
                         int*   __restrict__ clabel)      // [CAP]
{
    const int wave = (blockIdx.x * blockDim.x + threadIdx.x) >> 5;
    const int lane = threadIdx.x & 31;
    if (wave >= A) return;

    const float*  row  = cls + (size_t)wave * NCLS;
    const float2* row2 = (const float2*)row;              // 40 float2, 8B aligned

    // prefetch a future anchor's row into cache hierarchy (global_prefetch_b8)
    __builtin_prefetch(row + 32 * NCLS, 0, 0);

    // per-lane partial max/argmax (first-index wins ties)
    float2 v = row2[lane];                                // classes 2*lane, 2*lane+1
    float mv = v.x; int mc = 2 * lane;
    if (v.y > mv) { mv = v.y; mc = 2 * lane + 1; }
    if (lane < 8) {
        float2 v2 = row2[32 + lane];                      // classes 64+2*lane (+1)
        if (v2.x > mv) { mv = v2.x; mc = 64 + 2 * lane; }
        if (v2.y > mv) { mv = v2.y; mc = 64 + 2 * lane + 1; }
    }
    // wave32 shuffle reduction, prefer lower class index on exact tie
    #pragma unroll
    for (int off = 16; off >= 1; off >>= 1) {
        float ov = __shfl_xor(mv, off, 32);
        int   oc = __shfl_xor(mc, off, 32);
        if (ov > mv || (ov == mv && oc < mc)) { mv = ov; mc = oc; }
    }

    if (lane == 0) {
        float s = 1.0f / (1.0f + expf(-mv));              // sigmoid(max logit) == max(sigmoid)
        if (s > SCORE_THR) {
            // decode + clip only for candidates (~1% of anchors)
            const float4 a = *(const float4*)(anc + 4 * (size_t)wave); // y1,x1,y2,x2
            const float4 r = *(const float4*)(reg + 4 * (size_t)wave); // dy,dx,dh,dw
            float ya = (a.x + a.z) * 0.5f, xa = (a.y + a.w) * 0.5f;
            float ha = a.z - a.x,          wa = a.w - a.y;
            float w  = expf(r.w) * wa,     h  = expf(r.z) * ha;
            float yc = r.x * ha + ya,      xc = r.y * wa + xa;
            float x1 = fmaxf(xc - w * 0.5f, 0.0f);
            float y1 = fmaxf(yc - h * 0.5f, 0.0f);
            float x2 = fminf(xc + w * 0.5f, IMGF);
            float y2 = fminf(yc + h * 0.5f, IMGF);
            int slot = atomicAdd(cnt, 1);
            if (slot < CAP) {
                cscore[slot]       = s;
                cbox[slot * 4 + 0] = x1;
                cbox[slot * 4 + 1] = y1;
                cbox[slot * 4 + 2] = x2;
                cbox[slot * 4 + 3] = y2;
                clabel[slot]       = mc;
            }
        }
    }
}

// ---------------- kernel 2: greedy NMS (single workgroup) --------------
__global__ __launch_bounds__(NMS_THREADS)
void nms_kernel(const int* __restrict__ cnt,
                const float* __restrict__ cscore,
                const float* __restrict__ cbox,
                const int*   __restrict__ clabel,
                float* __restrict__ out)                  // 400 box + 100 score + 100 label
{
    __shared__ float s_score[CAP];                        // -1 == inactive (matches reference)
    __shared__ __align__(16) float s_box[CAP * 4];
    __shared__ float red_v[NMS_THREADS / 32];
    __shared__ int   red_i[NMS_THREADS / 32];
    __shared__ float s_pick[8];                           // box4, score, ok

    const int tid  = threadIdx.x;
    const int wv   = tid >> 5;
    const int lane = tid & 31;
    const int n    = min(*cnt, CAP);

    // ---- stage candidates into LDS (CDNA5 async global->LDS when available)
#if HAVE_ASYNC_LDS
    for (int j = tid; j < n; j += NMS_THREADS) {
        __builtin_amdgcn_global_load_async_to_lds_b32(
            (g_i32*)(cscore + j),
            (l_i32*)(&s_score[j]), 0, 0);
        __builtin_amdgcn_global_load_async_to_lds_b128(
            (g_v4i*)(cbox + 4 * j),
            (l_v4i*)(&s_box[4 * j]), 0, 0);
    }
    for (int j = tid; j < CAP; j += NMS_THREADS)
        if (j >= n) s_score[j] = -1.0f;
    __builtin_amdgcn_s_wait_asynccnt(0);
#else
    for (int j = tid; j < CAP; j += NMS_THREADS) {
        if (j < n) {
            s_score[j]     = cscore[j];
            float4 b       = *(const float4*)(cbox + 4 * j);
            s_box[4*j + 0] = b.x; s_box[4*j + 1] = b.y;
            s_box[4*j + 2] = b.z; s_box[4*j + 3] = b.w;
        } else {
            s_score[j] = -1.0f;
        }
    }
#endif
    __syncthreads();

    for (int it = 0; it < MAX_DET; ++it) {
        // ---- argmax over active candidates (2 slots/thread, wave reduce, 32-entry reduce)
        float bv = -3.0e38f; int bi = 0;
        for (int j = tid; j < CAP; j += NMS_THREADS) {
            float vv = s_score[j];
            if (vv > bv) { bv = vv; bi = j; }             // lower j wins ties (j ascending)
        }
        #pragma unroll
        for (int off = 16; off >= 1; off >>= 1) {
            float ov = __shfl_xor(bv, off, 32);
            int   oi = __shfl_xor(bi, off, 32);
            if (ov > bv || (ov == bv && oi < bi)) { bv = ov; bi = oi; }
        }
        if (lane == 0) { red_v[wv] = bv; red_i[wv] = bi; }
        __syncthreads();                                   // barrier A
        if (wv == 0) {
            float vv = red_v[lane]; int ii = red_i[lane];
            #pragma unroll
            for (int off = 16; off >= 1; off >>= 1) {
                float ov = __shfl_xor(vv, off, 32);
                int   oi = __shfl_xor(ii, off, 32);
                if (ov > vv || (ov == vv && oi < ii)) { vv = ov; ii = oi; }
            }
            if (lane == 0) {
                bool ok = vv > 0.0f;
                s_pick[5] = ok ? 1.0f : 0.0f;
                if (ok) {
                    float x1 = s_box[4*ii + 0], y1 = s_box[4*ii + 1];
                    float x2 = s_box[4*ii + 2], y2 = s_box[4*ii + 3];
                    s_pick[0] = x1; s_pick[1] = y1; s_pick[2] = x2; s_pick[3] = y2;
                    out[it * 4 + 0]       = x1;
                    out[it * 4 + 1]       = y1;
                    out[it * 4 + 2]       = x2;
                    out[it * 4 + 3]       = y2;
                    out[4 * MAX_DET + it] = vv;                      // score
                    out[5 * MAX_DET + it] = (float)clabel[ii];       // label
                }
            }
        }
        __syncthreads();                                   // barrier B
        if (s_pick[5] == 0.0f) break;                      // uniform: no candidates left

        // ---- suppress: active &= (iou < IOU_THR); picked box has iou==1 -> removed
        float px1 = s_pick[0], py1 = s_pick[1], px2 = s_pick[2], py2 = s_pick[3];
        float pa  = (px2 - px1) * (py2 - py1);
        for (int j = tid; j < CAP; j += NMS_THREADS) {
            float vv = s_score[j];
            if (vv > 0.0f) {
                float x1 = s_box[4*j + 0], y1 = s_box[4*j + 1];
                float x2 = s_box[4*j + 2], y2 = s_box[4*j + 3];
                float area  = (x2 - x1) * (y2 - y1);
                float ix1 = fmaxf(px1, x1), iy1 = fmaxf(py1, y1);
                float ix2 = fminf(px2, x2), iy2 = fminf(py2, y2);
                float inter = fmaxf(ix2 - ix1, 0.0f) * fmaxf(iy2 - iy1, 0.0f);
                float iou   = inter / (pa + area - inter + 1e-8f);
                if (!(iou < IOU_THR)) s_score[j] = -1.0f;
            }
        }
        __syncthreads();                                   // barrier C
    }
}

// ---------------- launcher ---------------------------------------------
extern "C" void kernel_launch(void* const* d_in, const int* in_sizes, int n_in,
                              void* d_out, int out_size, void* d_ws, size_t ws_size,
                              hipStream_t stream) {
    // setup_inputs order: x, regression, classification, anchors
    const float* reg = (const float*)d_in[1];
    const float* cls = (const float*)d_in[2];
    const float* anc = (const float*)d_in[3];
    const int A = in_sizes[1] / 4;                         // 110484

    // workspace layout
    char*  ws     = (char*)d_ws;
    int*   cnt    = (int*)ws;                              // counter @ 0
    float* cscore = (float*)(ws + 128);                    // CAP floats
    float* cbox   = cscore + CAP;                          // CAP*4 floats (16B aligned)
    int*   clabel = (int*)(cbox + CAP * 4);                // CAP ints
    (void)ws_size; (void)n_in;

    float* out = (float*)d_out;

    init_kernel<<<(out_size + 255) / 256, 256, 0, stream>>>(out, out_size, cnt);

    const int wavesPerBlock = 256 / 32;
    const int blocks = (A + wavesPerBlock - 1) / wavesPerBlock;
    score_decode_kernel<<<blocks, 256, 0, stream>>>(cls, reg, anc, A,
                                                    cnt, cscore, cbox, clabel);

    nms_kernel<<<1, NMS_THREADS, 0, stream>>>(cnt, cscore, cbox, clabel, out);
}